// ClassficationAndMDCA_31748398252041
// MI455X (gfx1250) — compile-verified
//
#include <hip/hip_runtime.h>
#include <hip/hip_bf16.h>

#define NCLS 128
#define NBLK 512
#define WAVES_PER_BLK 8
#define THREADS (WAVES_PER_BLK * 32)

typedef __attribute__((ext_vector_type(2))) float v2f;
typedef __attribute__((ext_vector_type(8))) float v8f;

// select x[i] for runtime i in [0,8) without spilling to scratch
__device__ __forceinline__ float pick8(const float x[8], int i) {
    float a01 = (i & 1) ? x[1] : x[0];
    float a23 = (i & 1) ? x[3] : x[2];
    float a45 = (i & 1) ? x[5] : x[4];
    float a67 = (i & 1) ? x[7] : x[6];
    float b0  = (i & 2) ? a23 : a01;
    float b1  = (i & 2) ? a67 : a45;
    return (i & 4) ? b1 : b0;
}

// Pass 1: streaming log-softmax + focal + per-class prob column-sums (WMMA).
// Layout per wave: lanes 0..15 process row r, lanes 16..31 process row r+1.
// Register v (v=0..7) of lane j holds class (j&15) + 16*v  -> this is exactly
// the B-matrix layout of V_WMMA_F32_16X16X4_F32 (4 rows x 16 cols per tile).
__global__ __launch_bounds__(THREADS) void fl_mdca_pass1(
    const float* __restrict__ inp, const long long* __restrict__ tgt,
    float* __restrict__ wsConf, int* __restrict__ wsCnt,
    float* __restrict__ wsFoc, float* __restrict__ wsNv, int nrows)
{
    __shared__ float sConf[WAVES_PER_BLK * NCLS];
    __shared__ int   sCnt[NCLS];
    __shared__ float sFoc[WAVES_PER_BLK];
    __shared__ float sNv[WAVES_PER_BLK];

    const int tid  = threadIdx.x;
    const int lane = tid & 31;
    const int wave = tid >> 5;
    const int l15  = lane & 15;
    const int half = lane >> 4;           // 0 = even row of pair, 1 = odd row

    for (int i = tid; i < NCLS; i += THREADS) sCnt[i] = 0;
    __syncthreads();

    v8f acc[8];
    #pragma unroll
    for (int v = 0; v < 8; ++v) acc[v] = (v8f){0.f,0.f,0.f,0.f,0.f,0.f,0.f,0.f};
    float focAcc = 0.f, nvAcc = 0.f;

    const int gw      = blockIdx.x * WAVES_PER_BLK + wave;  // wave-uniform
    const int nquads  = nrows >> 2;                         // 4 rows / iter
    const int wstride = NBLK * WAVES_PER_BLK;

    for (int q = gw; q < nquads; q += wstride) {
        const int r0 = q << 2;
        float p[2][8];

        #pragma unroll
        for (int pair = 0; pair < 2; ++pair) {
            const int row = r0 + 2 * pair + half;
            const float* rp = inp + (size_t)row * NCLS;
            float x[8];
            #pragma unroll
            for (int v = 0; v < 8; ++v) x[v] = rp[l15 + 16 * v];

            // per-half (16-lane) max reduction: two rows at once
            float m = x[0];
            #pragma unroll
            for (int v = 1; v < 8; ++v) m = fmaxf(m, x[v]);
            #pragma unroll
            for (int sh = 1; sh < 16; sh <<= 1) m = fmaxf(m, __shfl_xor(m, sh, 32));

            float e[8], s = 0.f;
            #pragma unroll
            for (int v = 0; v < 8; ++v) { e[v] = __expf(x[v] - m); s += e[v]; }
            #pragma unroll
            for (int sh = 1; sh < 16; sh <<= 1) s += __shfl_xor(s, sh, 32);

            const int   t     = (int)tgt[row];        // broadcast load per half
            const float maskf = (t != 0) ? 1.f : 0.f;
            const float inv   = __builtin_amdgcn_rcpf(s);
            const float scale = maskf * inv;
            #pragma unroll
            for (int v = 0; v < 8; ++v) p[pair][v] = e[v] * scale;

            // focal: logpt = x[t] - (m + log s);  source lane = half*16 + (t&15)
            const float logZ = m + __logf(s);
            const float xtl  = pick8(x, t >> 4);
            const float xt   = __shfl(xtl, (lane & 16) | (t & 15), 32);
            const float logpt = xt - logZ;
            const float pt    = __expf(logpt);
            const float f     = -(1.f - pt) * logpt * maskf;
            if (l15 == 0) {                  // one lane per row does bookkeeping
                focAcc += f;
                nvAcc  += maskf;
                if (t != 0) atomicAdd(&sCnt[t], 1);
            }
        }

        // D = ones(16x4) * B(4x16) + D : column sums of 4 rows x 16 classes.
        // A all-ones => result independent of K-row placement inside B.
        const v2f aones = {1.f, 1.f};
        #pragma unroll
        for (int v = 0; v < 8; ++v) {
            v2f b = { p[0][v], p[1][v] };
            acc[v] = __builtin_amdgcn_wmma_f32_16x16x4_f32(
                false, aones, false, b, (short)0, acc[v], false, false);
        }
    }

    // wave -> LDS: D row M=0 lives in vector element 0, lanes 0..15 (N = lane)
    if (lane < 16) {
        #pragma unroll
        for (int v = 0; v < 8; ++v)
            sConf[wave * NCLS + v * 16 + lane] = acc[v][0];
    }
    const float focC = focAcc + __shfl_xor(focAcc, 16, 32);
    const float nvC  = nvAcc  + __shfl_xor(nvAcc, 16, 32);
    if (lane == 0) { sFoc[wave] = focC; sNv[wave] = nvC; }
    __syncthreads();

    // block partials -> workspace (fixed-order, deterministic)
    if (tid < NCLS) {
        float c = 0.f;
        #pragma unroll
        for (int w = 0; w < WAVES_PER_BLK; ++w) c += sConf[w * NCLS + tid];
        wsConf[blockIdx.x * NCLS + tid] = c;
        wsCnt [blockIdx.x * NCLS + tid] = sCnt[tid];
    }
    if (tid == 0) {
        float f = 0.f, nv = 0.f;
        #pragma unroll
        for (int w = 0; w < WAVES_PER_BLK; ++w) { f += sFoc[w]; nv += sNv[w]; }
        wsFoc[blockIdx.x] = f;
        wsNv [blockIdx.x] = nv;
    }
}

// Pass 2: fixed-order reduction over block partials -> scalar loss
__global__ __launch_bounds__(128) void fl_mdca_pass2(
    const float* __restrict__ wsConf, const int* __restrict__ wsCnt,
    const float* __restrict__ wsFoc, const float* __restrict__ wsNv,
    float* __restrict__ out)
{
    __shared__ float red[NCLS];
    __shared__ float sScal[2];
    const int tid = threadIdx.x;

    float conf = 0.f, cnt = 0.f;
    for (int b = 0; b < NBLK; ++b) {
        conf += wsConf[b * NCLS + tid];
        cnt  += (float)wsCnt[b * NCLS + tid];
    }
    if (tid == 0) {
        float f = 0.f, nv = 0.f;
        for (int b = 0; b < NBLK; ++b) { f += wsFoc[b]; nv += wsNv[b]; }
        sScal[0] = f; sScal[1] = nv;
    }
    __syncthreads();
    const float nv = sScal[1];
    red[tid] = fabsf(conf - cnt) / nv;
    __syncthreads();
    for (int s = 64; s > 0; s >>= 1) {
        if (tid < s) red[tid] += red[tid + s];
        __syncthreads();
    }
    if (tid == 0) out[0] = sScal[0] / nv + red[0] * (1.0f / NCLS);
}

extern "C" void kernel_launch(void* const* d_in, const int* in_sizes, int n_in,
                              void* d_out, int out_size, void* d_ws, size_t ws_size,
                              hipStream_t stream) {
    const float*     inp = (const float*)d_in[0];
    const long long* tgt = (const long long*)d_in[1];
    const int nrows = in_sizes[1];          // N tokens (targets count)

    float* wsConf = (float*)d_ws;                    // NBLK * 128 floats
    int*   wsCnt  = (int*)(wsConf + NBLK * NCLS);    // NBLK * 128 ints
    float* wsFoc  = (float*)(wsCnt + NBLK * NCLS);   // NBLK floats
    float* wsNv   = wsFoc + NBLK;                    // NBLK floats

    fl_mdca_pass1<<<NBLK, THREADS, 0, stream>>>(inp, tgt, wsConf, wsCnt, wsFoc, wsNv, nrows);
    fl_mdca_pass2<<<1, 128, 0, stream>>>(wsConf, wsCnt, wsFoc, wsNv, (float*)d_out);
}